// GraphVAE_67027259621726
// MI455X (gfx1250) — compile-verified
//
#include <hip/hip_runtime.h>
#include <hip/hip_bf16.h>
#include <math.h>

// ---------------------------------------------------------------------------
// GraphVAE forward for MI455X (gfx1250, wave32).
// GEMMs use V_WMMA_F32_16X16X4_F32 (full f32 precision matrix pipe),
// one wave computes a 16x64 tile of C (A-fragment reused across 4 WMMAs).
// Irregular graph ops (degree, GCN scatter, mean pool) use f32 global atomics.
// ---------------------------------------------------------------------------

typedef __attribute__((ext_vector_type(2))) float v2f;
typedef __attribute__((ext_vector_type(8))) float v8f;

#define WAVES_PER_BLOCK 8   // 256 threads = 8 wave32s

// C[M,N] = act(A[M,K] @ B[K,N] + bias), A/B/C row-major.
// One wave computes one 16x64 tile of C. K must be a multiple of 4.
// OOB rows/cols are handled by clamping load indices (the polluted C entries
// are simply never stored), so the K-loop runs with full EXEC and no guards.
// act: 0 = none, 1 = relu, 2 = sigmoid
__global__ void wmma_gemm_f32(const float* __restrict__ A,
                              const float* __restrict__ B,
                              const float* __restrict__ bias,
                              float* __restrict__ C,
                              int M, int N, int K, int act) {
    const int lane   = threadIdx.x & 31;
    const int wave   = threadIdx.x >> 5;
    const int lane15 = lane & 15;
    const int tilesN = (N + 63) >> 6;                 // 64-wide C tiles
    const int tilesM = (M + 15) >> 4;
    const int tile = blockIdx.x * WAVES_PER_BLOCK + wave;
    if (tile >= tilesM * tilesN) return;              // wave-uniform exit
    const int tm = tile / tilesN;
    const int tn = tile % tilesN;
    const int m0 = tm << 4, n0 = tn << 6;

    // ISA 16x4 f32 A layout: lanes 0-15 hold K pair {0,1}; lanes 16-31 {2,3}
    const int kHalf = (lane >> 4) << 1;               // 0 or 2
    const int mA    = min(m0 + lane15, M - 1);        // clamped (memory-safe)

    int nB[4];
#pragma unroll
    for (int j = 0; j < 4; ++j)
        nB[j] = min(n0 + (j << 4) + lane15, N - 1);

    const float* ap = A + (size_t)mA * K + kHalf;
    const float* bp = B + (size_t)kHalf * N;

    __builtin_prefetch(ap, 0, 3);                     // global_prefetch_b8
    __builtin_prefetch(bp + nB[0], 0, 3);

    v8f c[4] = {};
#pragma unroll 4
    for (int k0 = 0; k0 < K; k0 += 4) {
        v2f a, b0, b1, b2, b3;
        a.x  = ap[0];        a.y  = ap[1];
        b0.x = bp[nB[0]];    b0.y = bp[N + nB[0]];
        b1.x = bp[nB[1]];    b1.y = bp[N + nB[1]];
        b2.x = bp[nB[2]];    b2.y = bp[N + nB[2]];
        b3.x = bp[nB[3]];    b3.y = bp[N + nB[3]];
        c[0] = __builtin_amdgcn_wmma_f32_16x16x4_f32(false, a, false, b0, (short)0, c[0], false, false);
        c[1] = __builtin_amdgcn_wmma_f32_16x16x4_f32(false, a, false, b1, (short)0, c[1], false, false);
        c[2] = __builtin_amdgcn_wmma_f32_16x16x4_f32(false, a, false, b2, (short)0, c[2], false, false);
        c[3] = __builtin_amdgcn_wmma_f32_16x16x4_f32(false, a, false, b3, (short)0, c[3], false, false);
        ap += 4;
        bp += (size_t)4 * N;
    }

    // C layout: VGPR v -> M = v + (lane>=16 ? 8 : 0), N = lane&15
    const int mhalf = (lane >> 4) << 3;
#pragma unroll
    for (int j = 0; j < 4; ++j) {
        const int n = n0 + (j << 4) + lane15;
        if (n < N) {
            const float bv = bias ? bias[n] : 0.0f;
#pragma unroll
            for (int v = 0; v < 8; ++v) {
                const int m = m0 + mhalf + v;
                if (m < M) {
                    float val = c[j][v] + bv;
                    if (act == 1)      val = fmaxf(val, 0.0f);
                    else if (act == 2) val = 1.0f / (1.0f + __expf(-val));
                    C[(size_t)m * N + n] = val;
                }
            }
        }
    }
}

// ---------------------------------------------------------------------------
__global__ void fill_f32(float* __restrict__ p, float v, int n) {
    int i = blockIdx.x * blockDim.x + threadIdx.x;
    if (i < n) p[i] = v;
}

__global__ void deg_edges(const int* __restrict__ dst, float* __restrict__ deg, int E) {
    int e = blockIdx.x * blockDim.x + threadIdx.x;
    if (e < E) atomicAdd(&deg[dst[e]], 1.0f);
}

__global__ void dinv_kernel(const float* __restrict__ deg, float* __restrict__ dinv, int n) {
    int i = blockIdx.x * blockDim.x + threadIdx.x;
    if (i < n) dinv[i] = rsqrtf(fmaxf(deg[i], 1.0f));
}

// out[i,f] = h[i,f] * dinv[i]^2   (self-loop contribution of GCN aggregation)
__global__ void self_loop_init(const float* __restrict__ h, const float* __restrict__ dinv,
                               float* __restrict__ out, int NN, int F) {
    int idx = blockIdx.x * blockDim.x + threadIdx.x;
    if (idx < NN * F) {
        int i = idx / F;
        float di = dinv[i];
        out[idx] = h[idx] * di * di;
    }
}

// out[dst[e], :] += h[src[e], :] * dinv[src]*dinv[dst]   (one block per edge)
__global__ void edge_scatter(const float* __restrict__ h,
                             const int* __restrict__ src, const int* __restrict__ dst,
                             const float* __restrict__ dinv,
                             float* __restrict__ out, int E, int F) {
    int e = blockIdx.x;
    if (e >= E) return;
    const int s = src[e], d = dst[e];
    const float coef = dinv[s] * dinv[d];
    const float4* hp = (const float4*)(h + (size_t)s * F);
    float* op = out + (size_t)d * F;
    for (int f4 = threadIdx.x; f4 < (F >> 2); f4 += blockDim.x) {
        float4 v = hp[f4];
        atomicAdd(op + f4 * 4 + 0, v.x * coef);
        atomicAdd(op + f4 * 4 + 1, v.y * coef);
        atomicAdd(op + f4 * 4 + 2, v.z * coef);
        atomicAdd(op + f4 * 4 + 3, v.w * coef);
    }
}

__global__ void bias_relu(float* __restrict__ x, const float* __restrict__ b, int NN, int F) {
    int idx = blockIdx.x * blockDim.x + threadIdx.x;
    if (idx < NN * F) x[idx] = fmaxf(x[idx] + b[idx % F], 0.0f);
}

// mean-pool accumulation: one block per node
__global__ void pool_acc(const float* __restrict__ h, const int* __restrict__ batch,
                         float* __restrict__ sums, float* __restrict__ cnt, int NN, int F) {
    int i = blockIdx.x;
    if (i >= NN) return;
    const int g = batch[i];
    const float* hp = h + (size_t)i * F;
    float* sp = sums + (size_t)g * F;
    for (int f = threadIdx.x; f < F; f += blockDim.x) atomicAdd(&sp[f], hp[f]);
    if (threadIdx.x == 0) atomicAdd(&cnt[g], 1.0f);
}

__global__ void pool_div(const float* __restrict__ sums, const float* __restrict__ cnt,
                         float* __restrict__ hg, int B, int F) {
    int idx = blockIdx.x * blockDim.x + threadIdx.x;
    if (idx < B * F) hg[idx] = sums[idx] / fmaxf(cnt[idx / F], 1.0f);
}

// z = mu + eps * exp(0.5*logvar)
__global__ void reparam(const float* __restrict__ mu, const float* __restrict__ lv,
                        const float* __restrict__ eps, float* __restrict__ z, int n) {
    int i = blockIdx.x * blockDim.x + threadIdx.x;
    if (i < n) z[i] = mu[i] + eps[i] * __expf(0.5f * lv[i]);
}

// adj[b,i,j] = adj[b,j,i] = probs[b, triu_index(i,j)]; diag = 0
__global__ void adj_build(const float* __restrict__ probs, float* __restrict__ adj,
                          int B, int MX, int P) {
    int idx = blockIdx.x * blockDim.x + threadIdx.x;
    const int total = B * MX * MX;
    if (idx >= total) return;
    const int b = idx / (MX * MX);
    const int r = idx % (MX * MX);
    const int i = r / MX;
    const int j = r % MX;
    float* ap = adj + (size_t)b * MX * MX;
    if (i == j) {
        ap[i * MX + j] = 0.0f;
    } else if (i < j) {
        const int p = i * (2 * MX - i - 1) / 2 + (j - i - 1);
        const float v = probs[(size_t)b * P + p];
        ap[i * MX + j] = v;
        ap[j * MX + i] = v;
    }
}

// ---------------------------------------------------------------------------
static inline int gemm_blocks(int M, int N) {
    int tiles = ((M + 15) >> 4) * ((N + 63) >> 6);
    return (tiles + WAVES_PER_BLOCK - 1) / WAVES_PER_BLOCK;
}

extern "C" void kernel_launch(void* const* d_in, const int* in_sizes, int n_in,
                              void* d_out, int out_size, void* d_ws, size_t ws_size,
                              hipStream_t stream) {
    constexpr int NN  = 20000;
    constexpr int E   = 320000;
    constexpr int NB  = 200;
    constexpr int FIN = 128;
    constexpr int H   = 256;
    constexpr int L   = 64;
    constexpr int MX  = 100;
    constexpr int P   = MX * (MX - 1) / 2;   // 4950

    const float* x    = (const float*)d_in[0];
    const int*   ei   = (const int*)d_in[1];
    const int*   src  = ei;
    const int*   dst  = ei + E;
    const int*   batch= (const int*)d_in[2];
    const float* eps  = (const float*)d_in[3];
    const float* W1   = (const float*)d_in[4];
    const float* b1   = (const float*)d_in[5];
    const float* W2   = (const float*)d_in[6];
    const float* b2   = (const float*)d_in[7];
    const float* Wmu  = (const float*)d_in[8];
    const float* bmu  = (const float*)d_in[9];
    const float* Wlv  = (const float*)d_in[10];
    const float* blv  = (const float*)d_in[11];
    const float* D1   = (const float*)d_in[12];
    const float* db1  = (const float*)d_in[13];
    const float* D2   = (const float*)d_in[14];
    const float* db2  = (const float*)d_in[15];
    const float* D3   = (const float*)d_in[16];
    const float* db3  = (const float*)d_in[17];

    // workspace layout (floats)
    float* ws    = (float*)d_ws;
    float* bufA  = ws;                          // NN*H   gemm outputs
    float* bufB  = bufA + (size_t)NN * H;       // NN*H   aggregation outputs
    float* deg   = bufB + (size_t)NN * H;       // NN
    float* dinv  = deg + NN;                    // NN
    float* sums  = dinv + NN;                   // NB*H
    float* cnt   = sums + (size_t)NB * H;       // NB
    float* hg    = cnt + NB;                    // NB*H
    float* z     = hg + (size_t)NB * H;         // NB*L
    float* dd1   = z + (size_t)NB * L;          // NB*H
    float* dd2   = dd1 + (size_t)NB * H;        // NB*H
    float* probs = dd2 + (size_t)NB * H;        // NB*P

    // outputs: (adj, mu, logvar) flattened
    float* adj = (float*)d_out;                 // NB*MX*MX
    float* mu  = adj + (size_t)NB * MX * MX;    // NB*L
    float* lv  = mu + (size_t)NB * L;           // NB*L

    // ---- degrees with self-loops, dinv = rsqrt(deg) ----
    fill_f32<<<(NN + 255) / 256, 256, 0, stream>>>(deg, 1.0f, NN);
    deg_edges<<<(E + 255) / 256, 256, 0, stream>>>(dst, deg, E);
    dinv_kernel<<<(NN + 255) / 256, 256, 0, stream>>>(deg, dinv, NN);

    // ---- GCN layer 1: h = relu(Agg(x @ W1) + b1) ----
    wmma_gemm_f32<<<gemm_blocks(NN, H), 256, 0, stream>>>(x, W1, nullptr, bufA, NN, H, FIN, 0);
    self_loop_init<<<(NN * H + 255) / 256, 256, 0, stream>>>(bufA, dinv, bufB, NN, H);
    edge_scatter<<<E, 64, 0, stream>>>(bufA, src, dst, dinv, bufB, E, H);
    bias_relu<<<(NN * H + 255) / 256, 256, 0, stream>>>(bufB, b1, NN, H);

    // ---- GCN layer 2: h = relu(Agg(h @ W2) + b2) ----
    wmma_gemm_f32<<<gemm_blocks(NN, H), 256, 0, stream>>>(bufB, W2, nullptr, bufA, NN, H, H, 0);
    self_loop_init<<<(NN * H + 255) / 256, 256, 0, stream>>>(bufA, dinv, bufB, NN, H);
    edge_scatter<<<E, 64, 0, stream>>>(bufA, src, dst, dinv, bufB, E, H);
    bias_relu<<<(NN * H + 255) / 256, 256, 0, stream>>>(bufB, b2, NN, H);

    // ---- global mean pool ----
    fill_f32<<<(NB * H + 255) / 256, 256, 0, stream>>>(sums, 0.0f, NB * H);
    fill_f32<<<(NB + 255) / 256, 256, 0, stream>>>(cnt, 0.0f, NB);
    pool_acc<<<NN, 256, 0, stream>>>(bufB, batch, sums, cnt, NN, H);
    pool_div<<<(NB * H + 255) / 256, 256, 0, stream>>>(sums, cnt, hg, NB, H);

    // ---- mu / logvar heads (written straight to d_out) ----
    wmma_gemm_f32<<<gemm_blocks(NB, L), 256, 0, stream>>>(hg, Wmu, bmu, mu, NB, L, H, 0);
    wmma_gemm_f32<<<gemm_blocks(NB, L), 256, 0, stream>>>(hg, Wlv, blv, lv, NB, L, H, 0);

    // ---- reparameterise ----
    reparam<<<(NB * L + 255) / 256, 256, 0, stream>>>(mu, lv, eps, z, NB * L);

    // ---- decoder MLP ----
    wmma_gemm_f32<<<gemm_blocks(NB, H), 256, 0, stream>>>(z,   D1, db1, dd1,   NB, H, L, 1);
    wmma_gemm_f32<<<gemm_blocks(NB, H), 256, 0, stream>>>(dd1, D2, db2, dd2,   NB, H, H, 1);
    wmma_gemm_f32<<<gemm_blocks(NB, P), 256, 0, stream>>>(dd2, D3, db3, probs, NB, P, H, 2);

    // ---- expand to symmetric adjacency ----
    adj_build<<<(NB * MX * MX + 255) / 256, 256, 0, stream>>>(probs, adj, NB, MX, P);
}